// OptimalModel_4312147165196
// MI455X (gfx1250) — compile-verified
//
#include <hip/hip_runtime.h>
#include <hip/hip_bf16.h>
#include <math.h>

// Problem constants (from reference): in [8,3,32,32], train [2048,3,32,32]
#define BATCH 8
#define MP    16          // M padded to WMMA tile height (rows 8..15 are zero)
#define DFEAT 3072        // 3*32*32
#define NTRAIN 2048

typedef __attribute__((ext_vector_type(2))) float v2f;
typedef __attribute__((ext_vector_type(8))) float v8f;
typedef __attribute__((ext_vector_type(4))) int   i32x4;

// gfx1250 async global->LDS copy path (ASYNCcnt), guarded so the file still
// compiles (direct-global fallback) on toolchains without the builtins.
#if defined(__has_builtin)
#if __has_builtin(__builtin_amdgcn_global_load_async_to_lds_b128) && \
    __has_builtin(__builtin_amdgcn_s_wait_asynccnt)
#define ATHENA_ASYNC_LDS 1
#endif
#endif

#if defined(ATHENA_ASYNC_LDS)
// builtin prototype (from hipcc diagnostic): (i32x4 AS1*, i32x4 AS3*, Ii, Ii)
__device__ __forceinline__ void async_cp_b128(const float* g, float* l) {
    __builtin_amdgcn_global_load_async_to_lds_b128(
        (__attribute__((address_space(1))) i32x4*)g,
        (__attribute__((address_space(3))) i32x4*)l,
        /*offset=*/0, /*cpol=*/0);
}
#endif

// ---------------------------------------------------------------------------
// scalar diffusion params, recomputed cheaply wherever needed (deterministic)
// ---------------------------------------------------------------------------
__device__ __forceinline__ void deno_params(const float* __restrict__ ng,
                                            float& alpha, float& sigma,
                                            float& vt, float& scale) {
    const float gamma    = ng[0] * 25.0f - 15.0f;          // [-15,10]
    const float variance = 1.0f / (1.0f + expf(gamma));    // sigmoid(-gamma)
    alpha = sqrtf(1.0f - variance);
    sigma = sqrtf(variance);
    vt    = expf(-gamma);
    scale = sqrtf(1.0f + vt);
}

// f32 WMMA 16x16x4: D = A(16x4) * B(4x16) + C  (one matrix per wave32)
__device__ __forceinline__ v8f wmma_f32_4(v2f a, v2f b, v8f c) {
    return __builtin_amdgcn_wmma_f32_16x16x4_f32(
        /*neg_a=*/false, a, /*neg_b=*/false, b,
        /*c_mod=*/(short)0, c, /*reuse_a=*/false, /*reuse_b=*/false);
}

// ---------------------------------------------------------------------------
// Kernel 0: state[m][k] = (m<8) ? in[m][k]*sqrt(1+vt)*mask[k]^2 : 0
//   (mask^2 folded into A so WMMA dot == sum(m^2 * s * t); the per-b
//    ||m*s||^2 term cancels in the softmax and is dropped entirely)
// ---------------------------------------------------------------------------
__global__ void prep_state_k(const float* __restrict__ in,
                             const float* __restrict__ mask,
                             const float* __restrict__ ng,
                             float* __restrict__ state) {
    int idx = blockIdx.x * blockDim.x + threadIdx.x;
    if (idx >= MP * DFEAT) return;
    int m = idx / DFEAT, k = idx - m * DFEAT;
    float alpha, sigma, vt, scale;
    deno_params(ng, alpha, sigma, vt, scale);
    float mv = mask[k];
    state[idx] = (m < BATCH) ? in[m * DFEAT + k] * scale * mv * mv : 0.0f;
}

// ---------------------------------------------------------------------------
// Kernel 1: tnorm[n] = sum_k (mask[k]*train[n][k])^2.  One wave per row.
// ---------------------------------------------------------------------------
__global__ void tnorm_k(const float* __restrict__ train,
                        const float* __restrict__ mask,
                        float* __restrict__ tnorm) {
    int wave = threadIdx.x >> 5, lane = threadIdx.x & 31;
    int n = blockIdx.x * 8 + wave;
    const float* row = train + (size_t)n * DFEAT;
    float s = 0.0f;
    for (int k = lane; k < DFEAT; k += 32) {
        float v = row[k] * mask[k];
        s = fmaf(v, v, s);
    }
    #pragma unroll
    for (int o = 16; o > 0; o >>= 1) s += __shfl_down(s, o);
    if (lane == 0) tnorm[n] = s;
}

// ---------------------------------------------------------------------------
// Kernel 2: dot[b][n] = state_row_b . train_row_n  via V_WMMA_F32_16X16X4_F32
//   Block = 8 waves; A (16x3072 state) staged once in LDS.
//   B tiles (16 rows x 64 K) double-buffered per wave via async global->LDS
//   (ASYNCcnt) when available; 16 WMMAs per chunk, 48 chunks.
// ---------------------------------------------------------------------------
#define ASTRIDE (DFEAT + 4)   // 3076 dwords -> bank-conflict-free, 16B aligned
#define KC      64            // K per chunk
#define BSTRIDE (KC + 4)      // 68 dwords per B-tile row
#define BTILE   (16 * BSTRIDE)
#define NCHUNK  (DFEAT / KC)  // 48

#if defined(ATHENA_ASYNC_LDS)
// one wave issues 8 x b128: copies its 16x64-float B tile (rows n0..n0+15)
__device__ __forceinline__ void async_b_tile(float* dst, const float* src,
                                             int lane) {
    #pragma unroll
    for (int j = 0; j < 8; ++j) {
        const int chunk = j * 32 + lane;        // 0..255 float4 chunks
        const int row   = chunk >> 4;           // 0..15  (training row)
        const int c4    = chunk & 15;           // 0..15  (float4 within row)
        async_cp_b128(src + (size_t)row * DFEAT + c4 * 4,
                      dst + row * BSTRIDE + c4 * 4);
    }
}
#endif

__global__ void dot_gemm_k(const float* __restrict__ state,
                           const float* __restrict__ train,
                           float* __restrict__ dot) {
    extern __shared__ float sA[];            // [16][ASTRIDE] then B tiles
    for (int i = threadIdx.x; i < MP * (DFEAT / 4); i += blockDim.x) {
        int row = i / (DFEAT / 4);
        int c4  = (i - row * (DFEAT / 4)) * 4;
        const float4 v = *(const float4*)(state + row * DFEAT + c4);
        *(float4*)(&sA[row * ASTRIDE + c4]) = v;
    }
    __syncthreads();

    const int wave = threadIdx.x >> 5, lane = threadIdx.x & 31;
    const int n0 = (blockIdx.x * 8 + wave) * 16;     // this wave's n-tile
    const int mcol  = lane & 15;                     // A row / B column
    const int khalf = (lane >> 4) * 2;               // K sub-pair per ISA layout
    const float* arow = &sA[mcol * ASTRIDE];

    v8f c = {};
#if defined(ATHENA_ASYNC_LDS)
    float* myB = &sA[MP * ASTRIDE + wave * 2 * BTILE];   // wave-private dbl buf
    const float* gbase = &train[(size_t)n0 * DFEAT];
    async_b_tile(myB, gbase, lane);                      // prime buffer 0
    const float* brow0 = myB + mcol * BSTRIDE;
    #pragma unroll 1
    for (int kc = 0; kc < NCHUNK; ++kc) {
        if (kc + 1 < NCHUNK) {
            async_b_tile(myB + ((kc + 1) & 1) * BTILE, gbase + (kc + 1) * KC,
                         lane);
            __builtin_amdgcn_s_wait_asynccnt(8);   // chunk kc landed in LDS
        } else {
            __builtin_amdgcn_s_wait_asynccnt(0);
        }
        const float* br = brow0 + (kc & 1) * BTILE;
        const float* ar = arow + kc * KC;
        #pragma unroll
        for (int kl = 0; kl < KC; kl += 4) {
            v2f a = *(const v2f*)(ar + kl + khalf);      // ds_load_b64
            v2f b = *(const v2f*)(br + kl + khalf);      // ds_load_b64
            c = wmma_f32_4(a, b, c);
        }
    }
#else
    const float* brow = &train[(size_t)(n0 + mcol) * DFEAT];
    #pragma unroll 4
    for (int k0 = 0; k0 < DFEAT; k0 += 4) {
        const int kk = k0 + khalf;
        v2f a = *(const v2f*)(arow + kk);                // ds_load_b64
        v2f b = *(const v2f*)(brow + kk);                // global_load_b64
        c = wmma_f32_4(a, b, c);
    }
#endif
    if (lane < 16) {                                  // lanes 0-15 hold M=0..7
        #pragma unroll
        for (int r = 0; r < BATCH; ++r)
            dot[r * NTRAIN + n0 + lane] = c[r];
    }
}

// ---------------------------------------------------------------------------
// Kernel 3: prob[b][n] = softmax_n( (dot - 0.5*tnorm)/vt ).
//   Blocks 0..7 do real rows; blocks 8..15 zero the WMMA padding rows.
// ---------------------------------------------------------------------------
__global__ void softmax_k(const float* __restrict__ dot,
                          const float* __restrict__ tnorm,
                          const float* __restrict__ ng,
                          float* __restrict__ prob) {
    const int b = blockIdx.x;
    if (b >= BATCH) {
        for (int n = threadIdx.x; n < NTRAIN; n += blockDim.x)
            prob[b * NTRAIN + n] = 0.0f;
        return;
    }
    float alpha, sigma, vt, scale;
    deno_params(ng, alpha, sigma, vt, scale);
    const float inv_vt = 1.0f / vt;

    const int wave = threadIdx.x >> 5, lane = threadIdx.x & 31;
    __shared__ float red[8];
    float lg[NTRAIN / 256];
    float lmax = -3.4e38f;
    #pragma unroll
    for (int i = 0; i < NTRAIN / 256; ++i) {
        int n = threadIdx.x + i * 256;
        float v = (dot[b * NTRAIN + n] - 0.5f * tnorm[n]) * inv_vt;
        lg[i] = v;
        lmax = fmaxf(lmax, v);
    }
    #pragma unroll
    for (int o = 16; o > 0; o >>= 1) lmax = fmaxf(lmax, __shfl_down(lmax, o));
    if (lane == 0) red[wave] = lmax;
    __syncthreads();
    float gmax = red[0];
    #pragma unroll
    for (int w = 1; w < 8; ++w) gmax = fmaxf(gmax, red[w]);
    __syncthreads();

    float lsum = 0.0f;
    #pragma unroll
    for (int i = 0; i < NTRAIN / 256; ++i) {
        lg[i] = expf(lg[i] - gmax);
        lsum += lg[i];
    }
    #pragma unroll
    for (int o = 16; o > 0; o >>= 1) lsum += __shfl_down(lsum, o);
    if (lane == 0) red[wave] = lsum;
    __syncthreads();
    float gsum = 0.0f;
    #pragma unroll
    for (int w = 0; w < 8; ++w) gsum += red[w];
    const float rnorm = 1.0f / gsum;
    #pragma unroll
    for (int i = 0; i < NTRAIN / 256; ++i)
        prob[b * NTRAIN + threadIdx.x + i * 256] = lg[i] * rnorm;
}

// ---------------------------------------------------------------------------
// Kernel 4: denoised = prob x train (WMMA over K=2048), fused epilogue
//   out = (in - alpha*denoised)/sigma * mask.  B rows are streamed with a
//   global prefetch 64 K-steps ahead (global_prefetch_b8).
// ---------------------------------------------------------------------------
#define PSTRIDE (NTRAIN + 4)   // 2052 dwords
__global__ void out_gemm_k(const float* __restrict__ prob,
                           const float* __restrict__ train,
                           const float* __restrict__ in,
                           const float* __restrict__ mask,
                           const float* __restrict__ ng,
                           float* __restrict__ out) {
    extern __shared__ float sP[];            // [16][PSTRIDE]
    for (int i = threadIdx.x; i < MP * (NTRAIN / 4); i += blockDim.x) {
        int row = i / (NTRAIN / 4);
        int c4  = (i - row * (NTRAIN / 4)) * 4;
        const float4 v = *(const float4*)(prob + row * NTRAIN + c4);
        *(float4*)(&sP[row * PSTRIDE + c4]) = v;
    }
    __syncthreads();

    const int wave = threadIdx.x >> 5, lane = threadIdx.x & 31;
    const int d0 = (blockIdx.x * 8 + wave) * 16;     // this wave's d-tile
    const int col   = lane & 15;
    const int khalf = (lane >> 4) * 2;
    const float* arow = &sP[col * PSTRIDE];
    const float* bcol = &train[d0 + col];

    v8f c = {};
    #pragma unroll 1
    for (int k0 = 0; k0 < NTRAIN; k0 += 16) {
        if (k0 + 64 < NTRAIN)                         // stay inside the buffer
            __builtin_prefetch(bcol + (size_t)(k0 + 64) * DFEAT, 0, 0);
        #pragma unroll
        for (int kl = 0; kl < 16; kl += 4) {
            const int kk = k0 + kl + khalf;
            v2f a = *(const v2f*)(arow + kk);         // ds_load_b64
            v2f b;
            b.x = bcol[(size_t)kk * DFEAT];           // coalesced across lanes
            b.y = bcol[(size_t)(kk + 1) * DFEAT];
            c = wmma_f32_4(a, b, c);
        }
    }

    float alpha, sigma, vt, scale;
    deno_params(ng, alpha, sigma, vt, scale);
    const float inv_sigma = 1.0f / sigma;
    if (lane < 16) {
        const int d = d0 + lane;
        const float mk = mask[d];
        #pragma unroll
        for (int r = 0; r < BATCH; ++r)
            out[r * DFEAT + d] = (in[r * DFEAT + d] - alpha * c[r]) * inv_sigma * mk;
    }
}

// ---------------------------------------------------------------------------
extern "C" void kernel_launch(void* const* d_in, const int* in_sizes, int n_in,
                              void* d_out, int out_size, void* d_ws, size_t ws_size,
                              hipStream_t stream) {
    const float* in    = (const float*)d_in[0];   // [8,3,32,32]
    const float* ng    = (const float*)d_in[1];   // [1]
    const float* mask  = (const float*)d_in[2];   // [3,32,32]
    const float* train = (const float*)d_in[3];   // [2048,3,32,32]
    float* out = (float*)d_out;

    // workspace layout (~401 KB total)
    float* ws_state = (float*)d_ws;                      // 16*3072
    float* ws_tnorm = ws_state + MP * DFEAT;             // 2048
    float* ws_dot   = ws_tnorm + NTRAIN;                 // 8*2048
    float* ws_prob  = ws_dot + BATCH * NTRAIN;           // 16*2048

    size_t lds_dot = (size_t)MP * ASTRIDE * sizeof(float);
#if defined(ATHENA_ASYNC_LDS)
    lds_dot += (size_t)8 * 2 * BTILE * sizeof(float);    // per-wave B dbl-buf
#endif

    prep_state_k<<<(MP * DFEAT + 255) / 256, 256, 0, stream>>>(in, mask, ng, ws_state);
    tnorm_k<<<NTRAIN / 8, 256, 0, stream>>>(train, mask, ws_tnorm);
    dot_gemm_k<<<NTRAIN / (8 * 16), 256, lds_dot, stream>>>(ws_state, train, ws_dot);
    softmax_k<<<MP, 256, 0, stream>>>(ws_dot, ws_tnorm, ng, ws_prob);
    out_gemm_k<<<DFEAT / (8 * 16), 256, MP * PSTRIDE * sizeof(float), stream>>>(
        ws_prob, train, in, mask, ng, out);
}